// GCNII_87978110091513
// MI455X (gfx1250) — compile-verified
//
#include <hip/hip_runtime.h>
#include <hip/hip_bf16.h>

#define N_NODES 50000
#define N_EDGES 600000
#define DIM 128
#define ALPHA_C 0.1f
#define BETA_C 0.22314355131420976f   // ln(1/4 + 1)
#define HS_STRIDE 132                 // padded h-strip row to avoid LDS bank conflicts

typedef __attribute__((ext_vector_type(2))) float v2f;
typedef __attribute__((ext_vector_type(8))) float v8f;

// -------- kernel 0: zero out accumulator (d_out) and degree buffer --------
__global__ void gcnii_zero(float* out, float* __restrict__ deg) {
    int idx = blockIdx.x * blockDim.x + threadIdx.x;
    if (idx < N_NODES * DIM) out[idx] = 0.0f;
    if (idx < N_NODES) deg[idx] = 0.0f;
}

// -------- kernel 1: in-degree via atomic add --------
__global__ void gcnii_degree(const int* __restrict__ dst, float* __restrict__ deg) {
    int e = blockIdx.x * blockDim.x + threadIdx.x;
    if (e < N_EDGES) atomicAdd(&deg[dst[e]], 1.0f);
}

// -------- kernel 2: norm = clip(deg,1)^-0.5 (in place) --------
__global__ void gcnii_norm(float* __restrict__ deg) {
    int i = blockIdx.x * blockDim.x + threadIdx.x;
    if (i < N_NODES) deg[i] = rsqrtf(fmaxf(deg[i], 1.0f));
}

// -------- kernel 3: edge scatter: agg[dst] += feat[src] * norm[src] --------
// one wave32 per edge; each lane handles 4 consecutive floats (float4 gather)
__global__ void gcnii_scatter(const float* __restrict__ feat,
                              const float* __restrict__ norm,
                              const int* __restrict__ src,
                              const int* __restrict__ dst,
                              float* agg) {
    int gtid = blockIdx.x * blockDim.x + threadIdx.x;
    int e    = gtid >> 5;
    int lane = gtid & 31;
    if (e >= N_EDGES) return;
    int s = src[e];
    int d = dst[e];
    float ns = norm[s];
    float4 v = *(const float4*)(feat + (size_t)s * DIM + lane * 4);
    float* ad = agg + (size_t)d * DIM + lane * 4;
    atomicAdd(ad + 0, v.x * ns);
    atomicAdd(ad + 1, v.y * ns);
    atomicAdd(ad + 2, v.z * ns);
    atomicAdd(ad + 3, v.w * ns);
}

// -------- kernel 4: fused blend + GEMM (WMMA f32 16x16x4) + identity map + bias --------
// block = 256 threads = 8 wave32s; block owns a 16-row strip of h;
// wave w computes output N-tile [w*16, w*16+16). W1 (64KB) staged in LDS.
__global__ __launch_bounds__(256) void gcnii_gemm(
    const float* agg,                  // aliases out (read strip before overwrite)
    const float* __restrict__ feat0,
    const float* __restrict__ W,
    const float* __restrict__ bias,
    const float* __restrict__ norm,
    float* out)
{
    __shared__ float Wl[DIM * DIM];          // 64 KB
    __shared__ float hs[16 * HS_STRIDE];     // 8.25 KB, padded stride
    __shared__ float bs[DIM];

    const int tid = threadIdx.x;
    const int m0  = blockIdx.x * 16;         // 50000 / 16 = 3125 exactly, no tail

    // stage W1 into LDS (float4, fully coalesced)
    for (int i = tid; i < DIM * DIM / 4; i += 256)
        ((float4*)Wl)[i] = ((const float4*)W)[i];
    if (tid < DIM) bs[tid] = bias[tid];

    // build h strip: h = agg*norm*(1-alpha) + alpha*feat_0
    for (int i = tid; i < 16 * DIM / 4; i += 256) {
        int r = i >> 5;              // row in strip (32 float4 per row)
        int c = (i & 31) * 4;
        int row = m0 + r;
        float nr = norm[row] * (1.0f - ALPHA_C);
        float4 a  = *(const float4*)(agg   + (size_t)row * DIM + c);
        float4 f0 = *(const float4*)(feat0 + (size_t)row * DIM + c);
        float4 h;
        h.x = a.x * nr + ALPHA_C * f0.x;
        h.y = a.y * nr + ALPHA_C * f0.y;
        h.z = a.z * nr + ALPHA_C * f0.z;
        h.w = a.w * nr + ALPHA_C * f0.w;
        *(float4*)(hs + r * HS_STRIDE + c) = h;
    }
    __syncthreads();

    const int wave = tid >> 5;
    const int lane = tid & 31;
    const int half = lane >> 4;      // 0: K pair {k0,k0+1}, 1: K pair {k0+2,k0+3}
    const int lr   = lane & 15;
    const int n0   = wave * 16;

    v8f acc = {};
    #pragma unroll
    for (int k0 = 0; k0 < DIM; k0 += 4) {
        const int ka = k0 + 2 * half;
        v2f a, b;
        // A fragment (16x4 f32): lane lr holds row M=lr, VGPR0=K=ka, VGPR1=K=ka+1
        a.x = hs[lr * HS_STRIDE + ka];
        a.y = hs[lr * HS_STRIDE + ka + 1];
        // B fragment (4x16 f32): lane lr holds col N=n0+lr, rows K=ka, ka+1
        b.x = Wl[ka * DIM + n0 + lr];
        b.y = Wl[(ka + 1) * DIM + n0 + lr];
        acc = __builtin_amdgcn_wmma_f32_16x16x4_f32(
            /*neg_a=*/false, a, /*neg_b=*/false, b,
            /*c_mod=*/(short)0, acc, /*reuse_a=*/false, /*reuse_b=*/false);
    }

    // D layout: VGPR r -> M = r (lanes 0-15) / r+8 (lanes 16-31), N = lr
    #pragma unroll
    for (int r = 0; r < 8; r++) {
        int lrow = r + 8 * half;
        int row  = m0 + lrow;
        int col  = n0 + lr;
        float hv = hs[lrow * HS_STRIDE + col];
        out[(size_t)row * DIM + col] =
            (1.0f - BETA_C) * hv + BETA_C * acc[r] + bs[col];
    }
}

extern "C" void kernel_launch(void* const* d_in, const int* in_sizes, int n_in,
                              void* d_out, int out_size, void* d_ws, size_t ws_size,
                              hipStream_t stream) {
    const float* feat  = (const float*)d_in[0];
    const float* feat0 = (const float*)d_in[1];
    const float* W     = (const float*)d_in[2];
    const float* bias  = (const float*)d_in[3];
    const int*   src   = (const int*)d_in[4];
    const int*   dst   = (const int*)d_in[5];
    float* out = (float*)d_out;
    float* deg = (float*)d_ws;   // N_NODES floats: degree -> norm (in place)

    gcnii_zero<<<(N_NODES * DIM + 255) / 256, 256, 0, stream>>>(out, deg);
    gcnii_degree<<<(N_EDGES + 255) / 256, 256, 0, stream>>>(dst, deg);
    gcnii_norm<<<(N_NODES + 255) / 256, 256, 0, stream>>>(deg);
    gcnii_scatter<<<(N_EDGES * 32 + 255) / 256, 256, 0, stream>>>(feat, deg, src, dst, out);
    gcnii_gemm<<<N_NODES / 16, 256, 0, stream>>>(out, feat0, W, bias, deg, out);
}